// MultiHeadEMA_53145925321336
// MI455X (gfx1250) — compile-verified
//
#include <hip/hip_runtime.h>
#include <hip/hip_bf16.h>

// MultiHeadEMA fused kernels for gfx1250 (MI455X).
// conv-with-geometric-kernel == N parallel 1st-order linear recurrences.
// Three-phase segmented scan for full-machine occupancy:
//   A: per-segment carry-out (zero carry-in), also warms L2 with x
//   B: sequential combine of segment carries (tiny)
//   C: fused chunked scan via V_WMMA_F32_16X16X4_F32 + gated-silu epilogue

#define SEQ_LEN 4096
#define BSZ     4
#define EMBED   1024
#define NDIM    2
#define TS      16                      // timestep block (WMMA M/N)
#define TILE_BLKS 4                     // blocks per tile -> 16 WMMA columns
#define TILE_L  (TS * TILE_BLKS)        // 64 timesteps per tile
#define NSEG    16                      // L-segments
#define SEG_L   (SEQ_LEN / NSEG)        // 256 timesteps per segment
#define SEG_TILES (SEG_L / TILE_L)      // 4 tiles per segment
#define DT      16                      // channels per workgroup
#define NPAIRS  (DT * NDIM)             // 32 (d,n) pairs per WG
#define WGSIZE  256
#define NCHAIN  (EMBED * NDIM * BSZ)    // 8192 carry chains

typedef float v2f __attribute__((ext_vector_type(2)));
typedef float v4f __attribute__((ext_vector_type(4)));
typedef float v8f __attribute__((ext_vector_type(8)));

__device__ __forceinline__ float sigmoidf_(float v) {
    return 1.0f / (1.0f + __expf(-v));
}

// ---------------------------------------------------------------------------
// Kernel A: per-segment carry-out with zero carry-in.
// grid (EMBED/DT, NSEG). ws layout: carry[seg][d][n][b]  (overwritten by B).
// ---------------------------------------------------------------------------
__global__ __launch_bounds__(WGSIZE)
void ema_seg_carry_kernel(const float* __restrict__ x,
                          const float* __restrict__ delta,
                          const float* __restrict__ alpha,
                          float* __restrict__ carry_ws)
{
    __shared__ __align__(16) float x_lds[TILE_L][BSZ][DT];
    __shared__ float qp_lds[DT][NDIM][TS + 1];

    const int tid = threadIdx.x;
    const int d0  = blockIdx.x * DT;
    const int seg = blockIdx.y;

    if (tid < NPAIRS) {
        const int dl = tid >> 1, n = tid & 1;
        const int gi = (d0 + dl) * NDIM + n;
        const float p = sigmoidf_(delta[gi]);
        const float q = 1.0f - p * sigmoidf_(alpha[gi]);
        float qq = 1.0f;
        for (int i = 0; i <= TS; ++i) { qp_lds[dl][n][i] = qq; qq *= q; }
    }

    // per-thread carry chain state (threads 0..127 own (dl,n,b) chains)
    const int dl = tid >> 3;
    const int n  = (tid >> 2) & 1;
    const int bb = tid & 3;
    float c = 0.0f;

    for (int t = 0; t < SEG_TILES; ++t) {
        const int l0 = seg * SEG_L + t * TILE_L;
        __syncthreads();
        {   // float4-coalesced stage of x[l0:l0+64, :, d0:d0+16]
            const int d4 = (tid & 3) << 2;
            const int r0 = tid >> 2;
            for (int k = 0; k < 4; ++k) {
                const int row = r0 + (k << 6);
                const int ll = row >> 2, b2 = row & 3;
                const v4f v = *reinterpret_cast<const v4f*>(
                    &x[((size_t)(l0 + ll) * BSZ + b2) * EMBED + d0 + d4]);
                *reinterpret_cast<v4f*>(&x_lds[ll][b2][d4]) = v;
            }
        }
        __syncthreads();
        if (tid < DT * NDIM * BSZ) {
            const float q16 = qp_lds[dl][n][TS];
            for (int blk = 0; blk < TILE_BLKS; ++blk) {
                float s = 0.0f;
                #pragma unroll
                for (int j = 0; j < TS; ++j)
                    s += qp_lds[dl][n][TS - 1 - j] * x_lds[blk * TS + j][bb][dl];
                c = q16 * c + s;
            }
        }
    }
    if (tid < DT * NDIM * BSZ)
        carry_ws[(((size_t)seg * EMBED + d0 + dl) * NDIM + n) * BSZ + bb] = c;
}

// ---------------------------------------------------------------------------
// Kernel B: turn per-segment carry-outs into carry-INs (sequential over NSEG).
// ---------------------------------------------------------------------------
__global__ __launch_bounds__(WGSIZE)
void ema_carry_combine_kernel(const float* __restrict__ delta,
                              const float* __restrict__ alpha,
                              float* __restrict__ carry_ws)
{
    const int gid = blockIdx.x * WGSIZE + threadIdx.x;   // 0..8191
    if (gid >= NCHAIN) return;
    const int d = gid >> 3;
    const int n = (gid >> 2) & 1;
    const int b = gid & 3;
    const int gi = d * NDIM + n;
    const float p = sigmoidf_(delta[gi]);
    const float q = 1.0f - p * sigmoidf_(alpha[gi]);
    float qs = q;                        // q^(2^k) -> q^256 after 8 squarings
    #pragma unroll
    for (int k = 0; k < 8; ++k) qs *= qs;

    float c = 0.0f;
    for (int s = 0; s < NSEG; ++s) {
        const size_t idx = (((size_t)s * EMBED + d) * NDIM + n) * BSZ + b;
        const float cout = carry_ws[idx];
        carry_ws[idx] = c;               // overwrite with carry-IN
        c = qs * c + cout;
    }
}

// ---------------------------------------------------------------------------
// Kernel C: fused WMMA chunked scan + gated silu. grid (EMBED/DT, NSEG).
// ---------------------------------------------------------------------------
__global__ __launch_bounds__(WGSIZE)
void ema_wmma_fused_kernel(const float* __restrict__ x,
                           const float* __restrict__ delta,
                           const float* __restrict__ alpha,
                           const float* __restrict__ beta,
                           const float* __restrict__ gamma_,
                           const float* __restrict__ omega,
                           const float* __restrict__ carry_ws,
                           float* __restrict__ out)
{
    __shared__ __align__(16) float x_lds[TILE_L][BSZ][DT];   // 16 KB
    __shared__ float s_lds[NPAIRS][TS][TS];                  // 32 KB
    __shared__ float qp_lds[DT][NDIM][TS + 1];
    __shared__ float w_lds[DT][NDIM];
    __shared__ float om_lds[DT];
    __shared__ float carry_lds[DT][NDIM][BSZ];
    __shared__ float cin_lds[DT][NDIM][TILE_BLKS][BSZ];

    const int tid   = threadIdx.x;
    const int lane  = tid & 31;
    const int wv    = tid >> 5;
    const int d0    = blockIdx.x * DT;
    const int seg   = blockIdx.y;

    // Phase 0: coefficients + segment carry-in
    if (tid < NPAIRS) {
        const int dl = tid >> 1, n = tid & 1;
        const int gi = (d0 + dl) * NDIM + n;
        const float p = sigmoidf_(delta[gi]);
        const float q = 1.0f - p * sigmoidf_(alpha[gi]);
        w_lds[dl][n] = p * beta[gi] * gamma_[gi] * 0.70710678118654752440f;
        float qq = 1.0f;
        for (int i = 0; i <= TS; ++i) { qp_lds[dl][n][i] = qq; qq *= q; }
        for (int b = 0; b < BSZ; ++b)
            carry_lds[dl][n][b] =
                carry_ws[(((size_t)seg * EMBED + d0 + dl) * NDIM + n) * BSZ + b];
    }
    if (tid < DT) om_lds[tid] = omega[d0 + tid];
    __syncthreads();

    // A-matrix operands in VGPRs: A[i][j] = (i>=j) ? q^(i-j) : 0.
    // f32 16x4 layout: lane%16 = M, K = vgprIdx + 2*(lane/16).
    const int m     = lane & 15;
    const int khalf = lane >> 4;
    v2f Amat[4][4];
    for (int pp = 0; pp < 4; ++pp) {
        const int pair = wv * 4 + pp;
        const int dl = pair >> 1, n = pair & 1;
        for (int k = 0; k < 4; ++k) {
            const int j0 = 4 * k + 2 * khalf;
            const float a0 = (m >= j0)     ? qp_lds[dl][n][m - j0]     : 0.0f;
            const float a1 = (m >= j0 + 1) ? qp_lds[dl][n][m - j0 - 1] : 0.0f;
            v2f av = {a0, a1};
            Amat[pp][k] = av;
        }
    }

    for (int t = 0; t < SEG_TILES; ++t) {
        const int l0 = seg * SEG_L + t * TILE_L;
        __syncthreads();

        {   // float4-coalesced stage (hits L2: kernel A already streamed x)
            const int d4 = (tid & 3) << 2;
            const int r0 = tid >> 2;
            for (int k = 0; k < 4; ++k) {
                const int row = r0 + (k << 6);
                const int ll = row >> 2, b2 = row & 3;
                const v4f v = *reinterpret_cast<const v4f*>(
                    &x[((size_t)(l0 + ll) * BSZ + b2) * EMBED + d0 + d4]);
                *reinterpret_cast<v4f*>(&x_lds[ll][b2][d4]) = v;
            }
        }
        if (t + 1 < SEG_TILES) {
            const float* nxt =
                x + ((size_t)(l0 + TILE_L + (tid >> 2)) * BSZ + (tid & 3)) * EMBED + d0;
            __builtin_prefetch(nxt, 0, 1);
        }
        __syncthreads();

        // WMMA: 4 (d,n) pairs per wave, 4 chained K=4 f32 WMMAs each.
        {
            const int col = lane & 15;
            const int blk = col >> 2;
            const int bb  = col & 3;
            for (int pp = 0; pp < 4; ++pp) {
                const int pair = wv * 4 + pp;
                const int dl = pair >> 1;
                v8f acc = {};
                for (int k = 0; k < 4; ++k) {
                    const int j0 = 4 * k + 2 * khalf;
                    v2f Bop = {x_lds[blk * TS + j0][bb][dl],
                               x_lds[blk * TS + j0 + 1][bb][dl]};
                    acc = __builtin_amdgcn_wmma_f32_16x16x4_f32(
                        false, Amat[pp][k], false, Bop, (short)0, acc, false, false);
                }
                for (int r = 0; r < 8; ++r)
                    s_lds[pair][r + 8 * khalf][col] = acc[r];
            }
        }
        __syncthreads();

        // Serial carry across the 4 blocks of this tile (128 chains).
        if (tid < DT * NDIM * BSZ) {
            const int dl = tid >> 3;
            const int n  = (tid >> 2) & 1;
            const int bb = tid & 3;
            const int pair = dl * NDIM + n;
            const float q16 = qp_lds[dl][n][TS];
            float c = carry_lds[dl][n][bb];
            for (int blk = 0; blk < TILE_BLKS; ++blk) {
                cin_lds[dl][n][blk][bb] = c;
                c = s_lds[pair][TS - 1][blk * 4 + bb] + q16 * c;
            }
            carry_lds[dl][n][bb] = c;
        }
        __syncthreads();

        // Epilogue: y = silu( sum_n w*(S + q^(i+1)*c_in) + x*omega ), b128 NT stores.
        {
            const int d4 = (tid & 3) << 2;
            const int r0 = tid >> 2;
            for (int k = 0; k < 4; ++k) {
                const int row = r0 + (k << 6);
                const int ll = row >> 2, b2 = row & 3;
                const int blk = ll >> 4;
                const int i   = ll & 15;
                const v4f xv = *reinterpret_cast<const v4f*>(&x_lds[ll][b2][d4]);
                v4f y;
                #pragma unroll
                for (int c4 = 0; c4 < 4; ++c4) {
                    const int dl = d4 + c4;
                    float acc = xv[c4] * om_lds[dl];
                    #pragma unroll
                    for (int n = 0; n < NDIM; ++n) {
                        const int pair = dl * NDIM + n;
                        const float s = s_lds[pair][i][blk * 4 + b2] +
                                        qp_lds[dl][n][i + 1] * cin_lds[dl][n][blk][b2];
                        acc += w_lds[dl][n] * s;
                    }
                    y[c4] = acc * sigmoidf_(acc);    // silu
                }
                __builtin_nontemporal_store(
                    y, reinterpret_cast<v4f*>(
                           &out[((size_t)(l0 + ll) * BSZ + b2) * EMBED + d0 + d4]));
            }
        }
    }
}

extern "C" void kernel_launch(void* const* d_in, const int* in_sizes, int n_in,
                              void* d_out, int out_size, void* d_ws, size_t ws_size,
                              hipStream_t stream) {
    (void)in_sizes; (void)n_in; (void)ws_size; (void)out_size;
    const float* x      = (const float*)d_in[0];
    const float* delta  = (const float*)d_in[1];
    const float* alpha  = (const float*)d_in[2];
    const float* beta   = (const float*)d_in[3];
    const float* gamma_ = (const float*)d_in[4];
    const float* omega  = (const float*)d_in[5];
    float* out      = (float*)d_out;
    float* carry_ws = (float*)d_ws;     // NSEG*EMBED*NDIM*BSZ floats = 512 KB

    dim3 blk(WGSIZE);
    dim3 gridA(EMBED / DT, NSEG);       // 1024 WGs
    ema_seg_carry_kernel<<<gridA, blk, 0, stream>>>(x, delta, alpha, carry_ws);

    dim3 gridB((NCHAIN + WGSIZE - 1) / WGSIZE);
    ema_carry_combine_kernel<<<gridB, blk, 0, stream>>>(delta, alpha, carry_ws);

    dim3 gridC(EMBED / DT, NSEG);       // 1024 WGs
    ema_wmma_fused_kernel<<<gridC, blk, 0, stream>>>(x, delta, alpha, beta,
                                                     gamma_, omega, carry_ws, out);
}